// TransformerConv_88218628260581
// MI455X (gfx1250) — compile-verified
//
#include <hip/hip_runtime.h>
#include <math.h>

// ---------------------------------------------------------------------------
// TransformerConv (e3nn attention) for MI455X / gfx1250, fp32 WMMA pipeline.
//   N=30000, F_IN=F_OUT=128, E=480000, Q=64, B=16, A=16, H=8
// Key identities used:
//   * edge length == 0 in the reference -> cutoff == exp(-0.1) (pos unused)
//   * score = q^T W_dot k  ==  f_dst (W_q @ W_dot) k  -> fold W_dot into W_q
//   * self connection == GEMM [N,2048] @ [2048,128] with A[n, i*16+j]=f_i*a_j
// ---------------------------------------------------------------------------

#define NN 30000
#define FIN 128
#define FOUT 128
#define EE 480000
#define QQ 64

typedef __attribute__((ext_vector_type(2))) float v2f;
typedef __attribute__((ext_vector_type(8))) float v8f;

// D = A(16x4) * B(4x16) + C, fp32 WMMA. A frag: lanes 0-15 hold K=k0,k0+1
// (VGPR0,VGPR1), lanes 16-31 hold K=k0+2,k0+3. B frag mirrored (row K striped
// across lanes within a VGPR).
__device__ __forceinline__ v8f wmma4(v2f a, v2f b, v8f c) {
  return __builtin_amdgcn_wmma_f32_16x16x4_f32(
      /*neg_a=*/false, a, /*neg_b=*/false, b,
      /*c_mod=*/(short)0, c, /*reuse_a=*/false, /*reuse_b=*/false);
}

// shifted softplus: log(1+exp(x)) - log(2), numerically stable
__device__ __forceinline__ float sspf(float x) {
  return fmaxf(x, 0.f) + log1pf(expf(-fabsf(x))) - 0.69314718055994531f;
}

// soft_unit_step(2*R_MAX*(1 - 0/R_MAX)) = exp(-1/10)
#define CUTOFF 0.90483741803595952f
#define INV_SQRT8 0.35355339059327373f

// --------------------------------------------------------------------------
__global__ __launch_bounds__(256) void k_zero(float* __restrict__ z) {
  int i = blockIdx.x * blockDim.x + threadIdx.x;
  if (i < NN) z[i] = 0.f;
}

// W_qd[128,64] = W_q[128,64] @ W_dot[64,64]  (one wave per 16x16 tile)
__global__ __launch_bounds__(32) void k_wqd(const float* __restrict__ Wq,
                                            const float* __restrict__ Wd,
                                            float* __restrict__ Wqd) {
  int lane = threadIdx.x & 31;
  int ln = lane & 15, half = lane >> 4, koff = half * 2;
  int mt = blockIdx.x >> 2, nt = blockIdx.x & 3;
  int col = nt * 16 + ln;
  v8f acc = {};
  for (int k0 = 0; k0 < 64; k0 += 4) {
    v2f a, b;
    a.x = Wq[(mt * 16 + ln) * 64 + k0 + koff];
    a.y = Wq[(mt * 16 + ln) * 64 + k0 + koff + 1];
    b.x = Wd[(k0 + koff) * 64 + col];
    b.y = Wd[(k0 + koff + 1) * 64 + col];
    acc = wmma4(a, b, acc);
  }
  for (int r = 0; r < 8; ++r)
    Wqd[(mt * 16 + r + half * 8) * 64 + col] = acc[r];
}

// qd[N,64] = node_feats[N,128] @ W_qd[128,64]; 1 block = 16 rows, 4 waves = 4 col tiles
__global__ __launch_bounds__(128) void k_qd(const float* __restrict__ nf,
                                            const float* __restrict__ Wqd,
                                            float* __restrict__ qd) {
  __shared__ float fs[16 * 132];
  int tid = threadIdx.x, lane = tid & 31, wid = tid >> 5;
  int m0 = blockIdx.x * 16;
  for (int t = 0; t < 16; ++t) {
    int idx = t * 128 + tid;
    int m = idx >> 7, f = idx & 127;
    fs[m * 132 + f] = nf[(m0 + m) * 128 + f];
  }
  __syncthreads();
  int ln = lane & 15, half = lane >> 4, koff = half * 2;
  int col = wid * 16 + ln;
  v8f acc = {};
  for (int k0 = 0; k0 < 128; k0 += 4) {
    v2f a, b;
    a.x = fs[ln * 132 + k0 + koff];
    a.y = fs[ln * 132 + k0 + koff + 1];
    b.x = Wqd[(k0 + koff) * 64 + col];
    b.y = Wqd[(k0 + koff + 1) * 64 + col];
    acc = wmma4(a, b, acc);
  }
  for (int r = 0; r < 8; ++r)
    qd[(m0 + r + half * 8) * 64 + col] = acc[r];
}

// out[N,128] = einsum('ni,nj,ijo->no') as [N,2048]@[2048,128], A built on the fly
__global__ __launch_bounds__(128) void k_selfconn(const float* __restrict__ nf,
                                                  const float* __restrict__ na,
                                                  const float* __restrict__ Wsc,
                                                  float* __restrict__ out) {
  __shared__ float fs[16 * 132];
  __shared__ float ats[16 * 17];
  int tid = threadIdx.x, lane = tid & 31, wid = tid >> 5;
  int m0 = blockIdx.x * 16;
  for (int t = 0; t < 16; ++t) {
    int idx = t * 128 + tid;
    int m = idx >> 7, f = idx & 127;
    fs[m * 132 + f] = nf[(m0 + m) * 128 + f];
  }
  {
    int m = tid >> 4, j = tid & 15;
    ats[m * 17 + j] = na[(m0 + m) * 16 + j];
    int idx2 = tid + 128;
    m = idx2 >> 4; j = idx2 & 15;
    ats[m * 17 + j] = na[(m0 + m) * 16 + j];
  }
  __syncthreads();
  int ln = lane & 15, half = lane >> 4, koff = half * 2;
  for (int ntp = 0; ntp < 2; ++ntp) {
    int nt = wid + ntp * 4;
    int col = nt * 16 + ln;
    v8f acc = {};
    for (int k0 = 0; k0 < 2048; k0 += 4) {
      int i = k0 >> 4;            // constant across the 4-wide K step
      int j = (k0 & 15) + koff;   // attr index
      float fv = fs[ln * 132 + i];
      v2f a, b;
      a.x = fv * ats[ln * 17 + j];
      a.y = fv * ats[ln * 17 + j + 1];
      b.x = Wsc[(k0 + koff) * 128 + col];
      b.y = Wsc[(k0 + koff + 1) * 128 + col];
      acc = wmma4(a, b, acc);
    }
    for (int r = 0; r < 8; ++r)
      out[(m0 + r + half * 8) * 128 + col] = acc[r];
  }
}

// Pass 1: per 16-edge tile per wave: radial_k, k = (x_src*attr*w_k)@W_k_lin,
// score = qd[dst]·k, exp -> expv, atomic z[dst]
__global__ __launch_bounds__(128) void k_pass1(
    const float* __restrict__ nf, const float* __restrict__ emb,
    const float* __restrict__ eattr, const int* __restrict__ ei,
    const float* __restrict__ W1k, const float* __restrict__ W2k,
    const float* __restrict__ Wkl, const float* __restrict__ qd,
    float* __restrict__ expv, float* __restrict__ zbuf) {
  __shared__ float xks[4][16 * 132];
  __shared__ float kms[4][16 * 68];
  __shared__ float hsh[4][16 * 8];
  __shared__ int sidx[4][16], didx[4][16];
  __shared__ float attrs[4][16];
  int tid = threadIdx.x, lane = tid & 31, wid = tid >> 5;
  int ge0 = (blockIdx.x * 4 + wid) * 16;

  if (lane < 16) {
    int e = lane, ge = ge0 + e;
    int s = ei[ge], d = ei[EE + ge];
    sidx[wid][e] = s;
    didx[wid][e] = d;
    attrs[wid][e] = eattr[ge];
    float em[16];
    for (int b = 0; b < 16; ++b) em[b] = emb[ge * 16 + b];
    for (int hh = 0; hh < 8; ++hh) {
      float accv = 0.f;
      for (int b = 0; b < 16; ++b) accv += em[b] * W1k[b * 8 + hh];
      hsh[wid][e * 8 + hh] = sspf(accv * 0.25f);  // W1 / sqrt(B=16)
    }
  }
  __syncthreads();
  // build xk[e][f] = x_src[f] * attr * w_k[f]; lanes sweep f, e = t>>2
  for (int t = 0; t < 64; ++t) {
    int e = t >> 2, f = (t & 3) * 32 + lane;
    float w = 0.f;
    for (int hh = 0; hh < 8; ++hh) w += hsh[wid][e * 8 + hh] * W2k[hh * 128 + f];
    w *= INV_SQRT8;  // W2 / sqrt(H=8)
    xks[wid][e * 132 + f] = nf[sidx[wid][e] * 128 + f] * attrs[wid][e] * w;
  }
  __syncthreads();
  int ln = lane & 15, half = lane >> 4, koff = half * 2;
  for (int nt = 0; nt < 4; ++nt) {
    int col = nt * 16 + ln;
    v8f acc = {};
    for (int k0 = 0; k0 < 128; k0 += 4) {
      v2f a, b;
      a.x = xks[wid][ln * 132 + k0 + koff];
      a.y = xks[wid][ln * 132 + k0 + koff + 1];
      b.x = Wkl[(k0 + koff) * 64 + col];
      b.y = Wkl[(k0 + koff + 1) * 64 + col];
      acc = wmma4(a, b, acc);
    }
    for (int r = 0; r < 8; ++r)
      kms[wid][(r + half * 8) * 68 + col] = acc[r];
  }
  __syncthreads();
  if (lane < 16) {
    int e = lane, ge = ge0 + e, d = didx[wid][e];
    float sc = 0.f;
    for (int q = 0; q < 64; ++q) sc += qd[d * 64 + q] * kms[wid][e * 68 + q];
    float ex = expf(sc) * CUTOFF;
    expv[ge] = ex;
    unsafeAtomicAdd(&zbuf[d], ex);
  }
}

// Pass 2: recompute v-side, scatter sqrt(alpha)*v into out with fp32 atomics
__global__ __launch_bounds__(128) void k_pass2(
    const float* __restrict__ nf, const float* __restrict__ emb,
    const float* __restrict__ eattr, const int* __restrict__ ei,
    const float* __restrict__ W1v, const float* __restrict__ W2v,
    const float* __restrict__ Wvl, const float* __restrict__ expv,
    const float* __restrict__ zbuf, float* __restrict__ out) {
  __shared__ float xvs[4][16 * 132];
  __shared__ float hsh[4][16 * 8];
  __shared__ int sidx[4][16], didx[4][16];
  __shared__ float attrs[4][16], coeff[4][16];
  int tid = threadIdx.x, lane = tid & 31, wid = tid >> 5;
  int ge0 = (blockIdx.x * 4 + wid) * 16;

  if (lane < 16) {
    int e = lane, ge = ge0 + e;
    int s = ei[ge], d = ei[EE + ge];
    sidx[wid][e] = s;
    didx[wid][e] = d;
    attrs[wid][e] = eattr[ge];
    float zz = zbuf[d];
    zz = (zz == 0.f) ? 1.f : zz;
    float al = expv[ge] / zz;
    coeff[wid][e] = sqrtf(fmaxf(al, 0.f));
    float em[16];
    for (int b = 0; b < 16; ++b) em[b] = emb[ge * 16 + b];
    for (int hh = 0; hh < 8; ++hh) {
      float accv = 0.f;
      for (int b = 0; b < 16; ++b) accv += em[b] * W1v[b * 8 + hh];
      hsh[wid][e * 8 + hh] = sspf(accv * 0.25f);
    }
  }
  __syncthreads();
  for (int t = 0; t < 64; ++t) {
    int e = t >> 2, f = (t & 3) * 32 + lane;
    float w = 0.f;
    for (int hh = 0; hh < 8; ++hh) w += hsh[wid][e * 8 + hh] * W2v[hh * 128 + f];
    w *= INV_SQRT8;
    xvs[wid][e * 132 + f] = nf[sidx[wid][e] * 128 + f] * attrs[wid][e] * w;
  }
  __syncthreads();
  int ln = lane & 15, half = lane >> 4, koff = half * 2;
  for (int nt = 0; nt < 8; ++nt) {
    int col = nt * 16 + ln;
    v8f acc = {};
    for (int k0 = 0; k0 < 128; k0 += 4) {
      v2f a, b;
      a.x = xvs[wid][ln * 132 + k0 + koff];
      a.y = xvs[wid][ln * 132 + k0 + koff + 1];
      b.x = Wvl[(k0 + koff) * 128 + col];
      b.y = Wvl[(k0 + koff + 1) * 128 + col];
      acc = wmma4(a, b, acc);
    }
    for (int r = 0; r < 8; ++r) {
      int e = r + half * 8;
      unsafeAtomicAdd(&out[didx[wid][e] * 128 + col], coeff[wid][e] * acc[r]);
    }
  }
}

// ---------------------------------------------------------------------------
extern "C" void kernel_launch(void* const* d_in, const int* in_sizes, int n_in,
                              void* d_out, int out_size, void* d_ws, size_t ws_size,
                              hipStream_t stream) {
  const float* node_feats = (const float*)d_in[0];
  const float* node_attrs = (const float*)d_in[1];
  const float* edge_emb   = (const float*)d_in[2];
  const float* edge_attrs = (const float*)d_in[3];
  // d_in[4] = pos — unused: edge length is identically 0 in the reference
  const int*   edge_index = (const int*)d_in[5];
  const float* W_q    = (const float*)d_in[6];
  const float* W1k    = (const float*)d_in[7];
  const float* W2k    = (const float*)d_in[8];
  const float* W1v    = (const float*)d_in[9];
  const float* W2v    = (const float*)d_in[10];
  const float* W_klin = (const float*)d_in[11];
  const float* W_vlin = (const float*)d_in[12];
  const float* W_dot  = (const float*)d_in[13];
  const float* W_sc   = (const float*)d_in[14];
  float* out = (float*)d_out;

  float* ws   = (float*)d_ws;
  float* Wqd  = ws;                 // 128*64
  float* qd   = Wqd + 128 * 64;     // NN*64
  float* expv = qd + (size_t)NN * 64;  // EE
  float* zbuf = expv + EE;          // NN    (total ~9.8 MB)

  k_zero<<<(NN + 255) / 256, 256, 0, stream>>>(zbuf);
  k_wqd<<<32, 32, 0, stream>>>(W_q, W_dot, Wqd);
  k_qd<<<NN / 16, 128, 0, stream>>>(node_feats, Wqd, qd);
  k_selfconn<<<NN / 16, 128, 0, stream>>>(node_feats, node_attrs, W_sc, out);
  k_pass1<<<EE / 64, 128, 0, stream>>>(node_feats, edge_emb, edge_attrs,
                                       edge_index, W1k, W2k, W_klin, qd, expv,
                                       zbuf);
  k_pass2<<<EE / 64, 128, 0, stream>>>(node_feats, edge_emb, edge_attrs,
                                       edge_index, W1v, W2v, W_vlin, expv,
                                       zbuf, out);
}